// Model_69767448756499
// MI455X (gfx1250) — compile-verified
//
#include <hip/hip_runtime.h>
#include <hip/hip_bf16.h>
#include <stdint.h>

// ---------------------------------------------------------------------------
// Top-K + Top-P (nucleus) filtering for sampling, MI455X / gfx1250.
//
// B=128 rows, V=128000 logits/row. One workgroup per row, 512 threads
// (16 wave32). Both thresholds are exact order statistics found by
// 3-round radix select on the monotonic uint32 key of the float logit:
//   - top-k:   count-weighted histogram select (k-th largest logit)
//   - top-p:   exp(x - max)-weighted histogram select (smallest prob p*
//              s.t. sum of probs >= p* is >= topP), Z from round-0 bins.
// Every pass streams the row through double-buffered LDS tiles using the
// CDNA5 async global->LDS path (ASYNCcnt + s_wait_asynccnt).
// ---------------------------------------------------------------------------

#define BATCH     128
#define VOCAB     128000
#define KCAP      1024
#define NTHREADS  512
#define TILE      2048           // floats per LDS tile = NTHREADS * 4
#define NTILES    ((VOCAB + TILE - 1) / TILE)   // 63 (last tile = 1024)

// 16-byte int vector type matching the builtin's parameter type
// (diagnostic: "__attribute__((__vector_size__(4 * sizeof(int)))) int *").
typedef int v4i_t __attribute__((vector_size(16)));

#if defined(__gfx1250__) && __has_builtin(__builtin_amdgcn_global_load_async_to_lds_b128)
#define USE_ASYNC_LDS 1
#else
#define USE_ASYNC_LDS 0
#if defined(__HIP_DEVICE_COMPILE__)
#warning "gfx1250 async global->LDS builtin unavailable; using synchronous LDS staging fallback"
#endif
#endif

#if USE_ASYNC_LDS
#if __has_builtin(__builtin_amdgcn_s_wait_asynccnt)
#define ASYNC_WAIT(n) __builtin_amdgcn_s_wait_asynccnt(n)
#else
#define ASYNC_WAIT(n) asm volatile("s_wait_asynccnt %0" ::"i"(n) : "memory")
#endif
#else
#define ASYNC_WAIT(n) ((void)0)
#endif

// Order-preserving float32 <-> uint32 key transform (larger key == larger float).
__device__ __forceinline__ unsigned f2ord(float f) {
  unsigned u = __float_as_uint(f);
  return (u & 0x80000000u) ? ~u : (u | 0x80000000u);
}
__device__ __forceinline__ float ord2f(unsigned o) {
  unsigned u = (o & 0x80000000u) ? (o & 0x7FFFFFFFu) : ~o;
  return __uint_as_float(u);
}

// Stage one tile (<= TILE floats) of the row into LDS. Every wave issues
// exactly ONE async instruction per tile (out-of-range threads clamp to the
// tile tail and redundantly copy the same 16B), so the per-wave ASYNCcnt is
// uniform and s_wait_asynccnt(1)/(0) implements exact double buffering.
__device__ __forceinline__ void stageTile(const float* __restrict__ row,
                                          float* __restrict__ ldsBuf,
                                          int tileBase, int tid) {
  int n = VOCAB - tileBase;
  if (n > TILE) n = TILE;             // n is always a multiple of 4, n >= 1024
  int off = tid * 4;
  if (off >= n) off = n - 4;          // clamp: duplicate copy of the tail 16B
#if USE_ASYNC_LDS
  __builtin_amdgcn_global_load_async_to_lds_b128(
      (v4i_t*)(row + tileBase + off),   // global source (16B)
      (v4i_t*)(ldsBuf + off),           // LDS destination (16B)
      0, 0);                            // imm offset, cpol
#else
  __builtin_prefetch(row + tileBase + off + TILE, 0, 0);  // global_prefetch_b8
  const float4 v = *reinterpret_cast<const float4*>(row + tileBase + off);
  *reinterpret_cast<float4*>(ldsBuf + off) = v;
#endif
}

// Stream the whole row through double-buffered LDS tiles; call f(i, value)
// for every element. Each thread processes the 4 contiguous floats it staged.
template <typename F>
__device__ __forceinline__ void rowForEach(const float* __restrict__ row,
                                           float* __restrict__ lds0,
                                           float* __restrict__ lds1,
                                           int tid, F&& f) {
  stageTile(row, lds0, 0, tid);
  for (int t = 0; t < NTILES; ++t) {
    float* cur = (t & 1) ? lds1 : lds0;
    if (t + 1 < NTILES) {
      stageTile(row, (t & 1) ? lds0 : lds1, (t + 1) * TILE, tid);
      ASYNC_WAIT(1);                  // tile t done (in-order async loads)
    } else {
      ASYNC_WAIT(0);
    }
    __syncthreads();                  // all waves' slices of tile t are in LDS
    const int base = t * TILE;
    const int n = (VOCAB - base < TILE) ? (VOCAB - base) : TILE;
    const int off = tid * 4;
    if (off < n) {
#pragma unroll
      for (int e = 0; e < 4; ++e) f(base + off + e, cur[off + e]);
    }
    __syncthreads();                  // tile t fully consumed before reuse
  }
}

__global__ __launch_bounds__(NTHREADS) void topkp_select_kernel(
    const float* __restrict__ logits, const long long* __restrict__ topK,
    const float* __restrict__ topP, long long* __restrict__ idxOut,
    float* __restrict__ selOut) {
  __shared__ __align__(16) float s_buf0[TILE];
  __shared__ __align__(16) float s_buf1[TILE];
  __shared__ unsigned s_hist[2048];           // aliased as float for pass D
  __shared__ unsigned s_chunkU[NTHREADS];
  __shared__ float s_chunkF[NTHREADS];
  __shared__ unsigned long long s_maxPack;
  __shared__ unsigned s_selBin, s_remU;
  __shared__ float s_remF, s_Z, s_W;
  __shared__ int s_crossed;

  const int b = blockIdx.x;
  const int tid = threadIdx.x;
  const float* row = logits + (size_t)b * VOCAB;

  const long long kq = topK[b];
  const bool kValid = (kq >= 1 && kq <= (long long)KCAP);
  const unsigned kTarget = kValid ? (unsigned)kq : 0u;
  const float P = topP[b];

  static const int kShift[3] = {21, 10, 0};
  static const int kBins[3] = {2048, 2048, 1024};
  static const unsigned kHiMask[3] = {0u, 0xFFE00000u, 0xFFFFFC00u};

  // ---- Pass A: row max + argmax (first index wins ties) -------------------
  if (tid == 0) s_maxPack = 0ull;
  __syncthreads();
  {
    unsigned long long best = 0ull;
    rowForEach(row, s_buf0, s_buf1, tid, [&](int i, float v) {
      unsigned long long p = ((unsigned long long)f2ord(v) << 32) |
                             (unsigned long long)(0xFFFFFFFFu - (unsigned)i);
      if (p > best) best = p;
    });
    atomicMax(&s_maxPack, best);
  }
  __syncthreads();
  const unsigned long long mp = s_maxPack;
  const float maxVal = ord2f((unsigned)(mp >> 32));
  const unsigned argmaxIdx = 0xFFFFFFFFu - (unsigned)(mp & 0xFFFFFFFFu);

  // ---- Pass B: radix select of the k-th largest key (count-weighted) ------
  unsigned kthKey = 0u;
  if (kValid) {
    unsigned prefix = 0u, target = kTarget;
    for (int r = 0; r < 3; ++r) {
      const int shift = kShift[r], bins = kBins[r];
      const unsigned hiMask = kHiMask[r], binMask = (unsigned)bins - 1u;
      for (int i = tid; i < bins; i += NTHREADS) s_hist[i] = 0u;
      __syncthreads();
      rowForEach(row, s_buf0, s_buf1, tid, [&](int, float v) {
        unsigned key = f2ord(v);
        if ((key & hiMask) == prefix)
          atomicAdd(&s_hist[(key >> shift) & binMask], 1u);
      });
      __syncthreads();
      const int nChunk = bins / 4;                 // chunked descending scan
      if (tid < nChunk) {
        int base = bins - 1 - 4 * tid;
        s_chunkU[tid] = s_hist[base] + s_hist[base - 1] + s_hist[base - 2] +
                        s_hist[base - 3];
      }
      __syncthreads();
      if (tid == 0) {
        unsigned cum = 0, sel = 0, rem = target;
        bool found = false;
        for (int c = 0; c < nChunk && !found; ++c) {
          unsigned ps = s_chunkU[c];
          if (cum + ps >= target) {
            for (int e = 0; e < 4; ++e) {
              int bi = bins - 1 - 4 * c - e;
              unsigned h = s_hist[bi];
              if (cum + h >= target) { sel = (unsigned)bi; rem = target - cum; found = true; break; }
              cum += h;
            }
          } else {
            cum += ps;
          }
        }
        s_selBin = sel; s_remU = rem;
      }
      __syncthreads();
      prefix |= s_selBin << shift;
      target = s_remU;
    }
    kthKey = prefix;  // exact key of the k-th largest logit; keep key >= kthKey
  }

  // ---- Pass D: radix select of nucleus threshold (exp-weighted) -----------
  float* histF = reinterpret_cast<float*>(s_hist);
  unsigned tKey = 0u;
  int crossed = 0;
  float Z = 1.0f, targetW = 0.0f;
  {
    unsigned prefixW = 0u;
    for (int r = 0; r < 3; ++r) {
      const int shift = kShift[r], bins = kBins[r];
      const unsigned hiMask = kHiMask[r], binMask = (unsigned)bins - 1u;
      for (int i = tid; i < bins; i += NTHREADS) histF[i] = 0.0f;
      __syncthreads();
      rowForEach(row, s_buf0, s_buf1, tid, [&](int, float v) {
        unsigned key = f2ord(v);
        if (kValid && key < kthKey) return;            // top-k mask
        if ((key & hiMask) != prefixW) return;
        atomicAdd(&histF[(key >> shift) & binMask], __expf(v - maxVal));
      });
      __syncthreads();
      const int nChunk = bins / 4;
      if (tid < nChunk) {
        int base = bins - 1 - 4 * tid;
        s_chunkF[tid] = histF[base] + histF[base - 1] + histF[base - 2] +
                        histF[base - 3];
      }
      __syncthreads();
      if (r == 0) {      // Z = total kept weight falls out of round-0 bins
        if (tid == 0) {
          float z = 0.0f;
          for (int c = 0; c < nChunk; ++c) z += s_chunkF[c];
          s_Z = z; s_W = P * z;
        }
        __syncthreads();
        Z = s_Z; targetW = s_W;
        __syncthreads();
      }
      if (tid == 0) {
        float cum = 0.0f, rem = 0.0f;
        int sel = -1;
        for (int c = 0; c < nChunk && sel < 0; ++c) {
          float ps = s_chunkF[c];
          if (cum + ps >= targetW) {
            for (int e = 0; e < 4; ++e) {     // re-check; on rounding miss,
              int bi = bins - 1 - 4 * c - e;  // cum is advanced and scan goes on
              float h = histF[bi];
              if (cum + h >= targetW) { sel = bi; rem = targetW - cum; break; }
              cum += h;
            }
          } else {
            cum += ps;
          }
        }
        s_selBin = (unsigned)sel; s_remF = rem; s_crossed = (sel >= 0) ? 1 : 0;
      }
      __syncthreads();
      if (!s_crossed) { crossed = 0; break; }          // uniform across block
      crossed = 1;
      prefixW |= s_selBin << shift;
      targetW = s_remF;
    }
    tKey = prefixW;  // exact key at which the descending prob cumsum crosses P
  }

  // ---- Pass E: emit filtered logits + argmax index ------------------------
  const float thr = crossed ? (__expf(ord2f(tKey) - maxVal) / Z) : 0.0f;
  float* outRow = selOut + (size_t)b * VOCAB;
  const float NEG_INF = __int_as_float((int)0xFF800000u);
  const bool kv = kValid;
  const unsigned kk = kthKey;
  rowForEach(row, s_buf0, s_buf1, tid, [&](int i, float v) {
    unsigned key = f2ord(v);
    bool kept = (!kv) || (key >= kk);                  // top-k mask
    float p = kept ? (__expf(v - maxVal) / Z) : 0.0f;  // softmax prob
    outRow[i] = (p >= thr && kept) ? v : NEG_INF;      // nucleus mask
  });

  if (tid == 0) idxOut[b] = (long long)argmaxIdx;
}

// ---------------------------------------------------------------------------
// Host launch. Inputs (setup_inputs order):
//   [0] logits f32 [B,V]   [1] topK i64 [B]   [2] topP f32 [B]
//   [3] q (unused) [4] eps (unused) [5] isNeedLogits (unused) [6] topKGuess (unused)
// Output tuple flat-concatenated: int64 idx[B] (== 2*B float slots), then
// float32 logitsTopKPSelect[B,V].
// ---------------------------------------------------------------------------
extern "C" void kernel_launch(void* const* d_in, const int* in_sizes, int n_in,
                              void* d_out, int out_size, void* d_ws,
                              size_t ws_size, hipStream_t stream) {
  (void)in_sizes; (void)n_in; (void)out_size; (void)d_ws; (void)ws_size;
  const float* logits = (const float*)d_in[0];
  const long long* topK = (const long long*)d_in[1];
  const float* topP = (const float*)d_in[2];
  long long* idxOut = (long long*)d_out;
  float* selOut = (float*)d_out + 2 * BATCH;  // int64[B] occupies 2*B f32 slots
  topkp_select_kernel<<<dim3(BATCH), dim3(NTHREADS), 0, stream>>>(
      logits, topK, topP, idxOut, selOut);
}